// GCN_84189948936514
// MI455X (gfx1250) — compile-verified
//
#include <hip/hip_runtime.h>
#include <hip/hip_bf16.h>

#define NN 20000
#define EE 640000
#define DD 128

typedef float v2f __attribute__((ext_vector_type(2)));
typedef float v8f __attribute__((ext_vector_type(8)));

__global__ __launch_bounds__(256) void zero_f32(float* __restrict__ p, int n) {
  int i = blockIdx.x * blockDim.x + threadIdx.x;
  if (i < n) p[i] = 0.0f;
}

__global__ __launch_bounds__(256) void degree_count(const long long* __restrict__ dst,
                                                    float* __restrict__ deg, int e) {
  int i = blockIdx.x * blockDim.x + threadIdx.x;
  if (i < e) atomicAdd(&deg[(int)dst[i]], 1.0f);
}

__global__ __launch_bounds__(256) void degree_inv(float* __restrict__ deg, int n) {
  int i = blockIdx.x * blockDim.x + threadIdx.x;
  if (i < n) { float d = deg[i]; deg[i] = (d > 0.0f) ? (1.0f / d) : 0.0f; }
}

// xt = x @ W^T + b  via V_WMMA_F32_16X16X4_F32.
// grid.x = N/16 M-tiles; 256 threads = 8 waves; wave w owns N-tile w (D=128 -> 8 tiles).
// A (16x4 f32): lanes 0-15 K=0,1; lanes 16-31 K=2,3 -> float2 load at k+2*half.
// B (4x16):     B[k][n] = W[n][k], mirrored layout.
// C/D (16x16):  VGPR v -> row v+8*half, col = lane&15. Bias broadcast per column.
__global__ __launch_bounds__(256) void gemm_bias_wmma(const float* __restrict__ x,
                                                      const float* __restrict__ W,
                                                      const float* __restrict__ bias,
                                                      float* __restrict__ xt) {
  const int wave = threadIdx.x >> 5;
  const int lane = threadIdx.x & 31;
  const int half = lane >> 4;
  const int lq   = lane & 15;
  const int m0 = blockIdx.x << 4;
  const int n0 = wave << 4;

  const float* arow = x + (size_t)(m0 + lq) * DD;
  const float* brow = W + (size_t)(n0 + lq) * DD;

  v8f c;
  const float bv = bias[n0 + lq];
#pragma unroll
  for (int i = 0; i < 8; ++i) c[i] = bv;

#pragma unroll 8
  for (int k = 0; k < DD; k += 4) {
    v2f a = *(const v2f*)(arow + k + 2 * half);
    v2f b = *(const v2f*)(brow + k + 2 * half);
    c = __builtin_amdgcn_wmma_f32_16x16x4_f32(false, a, false, b, (short)0, c,
                                              false, false);
  }

  float* orow = xt + (size_t)(m0 + 8 * half) * DD + n0 + lq;
#pragma unroll
  for (int v = 0; v < 8; ++v) orow[(size_t)v * DD] = c[v];
}

// One wave32 per edge; each lane moves 4 consecutive floats (512B coalesced per wave).
// agg stays L2-resident (10 MB << 192 MB L2), so atomics RMW on-chip.
__global__ __launch_bounds__(256) void scatter_sum(const float* __restrict__ xt,
                                                   const long long* __restrict__ src,
                                                   const long long* __restrict__ dst,
                                                   float* __restrict__ agg) {
  const int e = blockIdx.x * 8 + (threadIdx.x >> 5);
  const int lane = threadIdx.x & 31;
  if (e >= EE) return;
  const int s = (int)src[e];
  const int d = (int)dst[e];
  const float4 v = *(const float4*)(xt + (size_t)s * DD + lane * 4);
  float* p = agg + (size_t)d * DD + lane * 4;
  atomicAdd(p + 0, v.x);
  atomicAdd(p + 1, v.y);
  atomicAdd(p + 2, v.z);
  atomicAdd(p + 3, v.w);
}

// out = relu(agg * deg_inv) + xt, float4-vectorized.
__global__ __launch_bounds__(256) void finalize(const float* __restrict__ agg,
                                                const float* __restrict__ xt,
                                                const float* __restrict__ deginv,
                                                float* __restrict__ out) {
  const int t = blockIdx.x * blockDim.x + threadIdx.x;
  if (t >= NN * (DD / 4)) return;
  const int n = t >> 5;
  const int cOff = (t & 31) * 4;
  const float di = deginv[n];
  const float4 a  = *(const float4*)(agg + (size_t)n * DD + cOff);
  const float4 xv = *(const float4*)(xt + (size_t)n * DD + cOff);
  float4 r;
  r.x = fmaxf(a.x * di, 0.0f) + xv.x;
  r.y = fmaxf(a.y * di, 0.0f) + xv.y;
  r.z = fmaxf(a.z * di, 0.0f) + xv.z;
  r.w = fmaxf(a.w * di, 0.0f) + xv.w;
  *(float4*)(out + (size_t)n * DD + cOff) = r;
}

extern "C" void kernel_launch(void* const* d_in, const int* in_sizes, int n_in,
                              void* d_out, int out_size, void* d_ws, size_t ws_size,
                              hipStream_t stream) {
  (void)in_sizes; (void)n_in; (void)out_size; (void)ws_size;

  const float*      x  = (const float*)d_in[0];
  const long long*  ei = (const long long*)d_in[1];   // int64 [2, E]
  const float* Wts[3] = {(const float*)d_in[2], (const float*)d_in[4], (const float*)d_in[6]};
  const float* bs[3]  = {(const float*)d_in[3], (const float*)d_in[5], (const float*)d_in[7]};
  const long long* src = ei;
  const long long* dst = ei + EE;

  float* ws = (float*)d_ws;
  const size_t ND = (size_t)NN * DD;
  float* xt   = ws;            // [N, D]
  float* agg  = ws + ND;       // [N, D]
  float* xbuf = ws + 2 * ND;   // [N, D] layer ping buffer
  float* deg  = ws + 3 * ND;   // [N]  (deg -> deg_inv in place)

  // degree -> inverse degree (deg_inv = deg>0 ? 1/deg : 0)
  zero_f32<<<(NN + 255) / 256, 256, 0, stream>>>(deg, NN);
  degree_count<<<(EE + 255) / 256, 256, 0, stream>>>(dst, deg, EE);
  degree_inv<<<(NN + 255) / 256, 256, 0, stream>>>(deg, NN);

  const float* cur = x;
  for (int l = 0; l < 3; ++l) {
    float* nxt = (l == 2) ? (float*)d_out : xbuf;
    gemm_bias_wmma<<<NN / 16, 256, 0, stream>>>(cur, Wts[l], bs[l], xt);
    zero_f32<<<((int)ND + 255) / 256, 256, 0, stream>>>(agg, (int)ND);
    scatter_sum<<<EE / 8, 256, 0, stream>>>(xt, src, dst, agg);
    finalize<<<(NN * (DD / 4) + 255) / 256, 256, 0, stream>>>(agg, xt, deg, nxt);
    cur = nxt;
  }
}